// SAGE_18554258719492
// MI455X (gfx1250) — compile-verified
//
#include <hip/hip_runtime.h>
#include <hip/hip_bf16.h>

// ---------------------------------------------------------------------------
// GraphSAGE 2-layer forward for MI455X (gfx1250).
//   layer: h = relu( (mean-agg of x at dst) @ W_l + b + x @ W_r )
// Aggregation: f32 global atomics (L2-resident, 51.2MB per feature matrix).
// GEMM: V_WMMA_F32_16X16X4_F32 (fp32-exact), K=128 in steps of 4.
// ---------------------------------------------------------------------------

typedef __attribute__((ext_vector_type(2))) float v2f;
typedef __attribute__((ext_vector_type(8))) float v8f;

#define N_NODES 100000
#define N_EDGES 1600000
#define D       128

// ---------------------------- zero fill ------------------------------------
__global__ __launch_bounds__(256) void zero_f4(float4* p, int n4) {
    int i = blockIdx.x * 256 + threadIdx.x;
    if (i < n4) p[i] = make_float4(0.f, 0.f, 0.f, 0.f);
}

__global__ __launch_bounds__(256) void zero_f(float* p, int n) {
    int i = blockIdx.x * 256 + threadIdx.x;
    if (i < n) p[i] = 0.f;
}

// ---------------------------- degree ---------------------------------------
__global__ __launch_bounds__(256) void degree_kernel(const long long* __restrict__ dst,
                                                     float* __restrict__ deg, int E) {
    int e = blockIdx.x * 256 + threadIdx.x;
    if (e < E) atomicAdd(deg + (int)dst[e], 1.0f);
}

__global__ __launch_bounds__(256) void recip_kernel(float* __restrict__ deg, int n) {
    int i = blockIdx.x * 256 + threadIdx.x;
    if (i < n) deg[i] = 1.0f / fmaxf(deg[i], 1.0f);
}

// ---------------------------- edge scatter-add -----------------------------
// One wave covers one edge: 32 lanes x float4 = 128 features, fully coalesced
// 512B gather from src row and 512B of f32 atomics into dst row. Edge-index
// loads are wave-uniform (scalar path).
__global__ __launch_bounds__(256) void scatter_add_kernel(
        const float* __restrict__ feat,
        const long long* __restrict__ src_idx,
        const long long* __restrict__ dst_idx,
        float* __restrict__ agg, int E) {
    long long i = (long long)blockIdx.x * 256 + threadIdx.x;   // E*32 threads
    int e = (int)(i >> 5);
    int q = (int)(i & 31);
    if (e >= E) return;
    int s = (int)src_idx[e];
    int d = (int)dst_idx[e];
    const float4 v = *(const float4*)(feat + (long long)s * D + q * 4);
    float* o = agg + (long long)d * D + q * 4;
    atomicAdd(o + 0, v.x);
    atomicAdd(o + 1, v.y);
    atomicAdd(o + 2, v.z);
    atomicAdd(o + 3, v.w);
}

// ---------------------------- fused SAGE GEMM ------------------------------
// out[m, :] = (rdeg[m] * agg[m, :]) @ Wl + b + x[m, :] @ Wr   (+ optional ReLU)
// Block = 256 threads = 8 waves; block -> one 16-row tile; wave w -> 16-col
// tile w (8 * 16 = 128 output cols). Each wave runs 64 v_wmma_f32_16x16x4_f32
// (32 K-steps x 2 input matrices) into one v8f accumulator.
//
// ISA operand layouts (CDNA5 ISA):
//   A 16x4 f32:  lane<16 holds A[lane][k], A[lane][k+1]; lane>=16 holds K=k+2,k+3
//   B 4x16 f32:  v0 = B[k + 2*half][col], v1 = B[k+1 + 2*half][col]
//   C 16x16 f32: VGPR i -> M = i + 8*half, N = n0 + (lane & 15)
template <bool RELU>
__global__ __launch_bounds__(256) void sage_gemm_kernel(
        const float* __restrict__ agg, const float* __restrict__ xin,
        const float* __restrict__ rdeg,
        const float* __restrict__ Wl, const float* __restrict__ bias,
        const float* __restrict__ Wr, float* __restrict__ out) {
    const int lane = threadIdx.x & 31;
    const int wave = threadIdx.x >> 5;     // n-tile 0..7
    const int half = lane >> 4;            // 0|1 (K split for A, B)
    const int lm   = lane & 15;
    const int m0   = blockIdx.x * 16;      // 100000 % 16 == 0: all tiles full
    const int row  = m0 + lm;
    const int col  = wave * 16 + lm;

    const float rd = rdeg[row];
    const float bv = bias[col];
    v8f c;
#pragma unroll
    for (int i = 0; i < 8; ++i) c[i] = bv;

    const float* ap = agg + (long long)row * D;
    const float* xp = xin + (long long)row * D;

#pragma unroll 4
    for (int k = 0; k < D; k += 4) {
        const int ka = k + 2 * half;
        v2f a_agg, a_x, b_l, b_r;
        a_agg[0] = ap[ka] * rd;  a_agg[1] = ap[ka + 1] * rd;
        a_x[0]   = xp[ka];       a_x[1]   = xp[ka + 1];
        b_l[0] = Wl[ka * D + col];  b_l[1] = Wl[(ka + 1) * D + col];
        b_r[0] = Wr[ka * D + col];  b_r[1] = Wr[(ka + 1) * D + col];
        c = __builtin_amdgcn_wmma_f32_16x16x4_f32(false, a_agg, false, b_l,
                                                  (short)0, c, false, false);
        c = __builtin_amdgcn_wmma_f32_16x16x4_f32(false, a_x, false, b_r,
                                                  (short)0, c, false, false);
    }

#pragma unroll
    for (int i = 0; i < 8; ++i) {
        float v = c[i];
        if (RELU) v = fmaxf(v, 0.f);
        out[(long long)(m0 + half * 8 + i) * D + wave * 16 + lm] = v;
    }
}

// ---------------------------------------------------------------------------
extern "C" void kernel_launch(void* const* d_in, const int* in_sizes, int n_in,
                              void* d_out, int out_size, void* d_ws, size_t ws_size,
                              hipStream_t stream) {
    // setup_inputs() order:
    const float*     x    = (const float*)d_in[0];              // [N, 128]
    const long long* ei   = (const long long*)d_in[1];          // [2, E] int64
    const float*     W1l  = (const float*)d_in[2];
    const float*     b1   = (const float*)d_in[3];
    const float*     W1r  = (const float*)d_in[4];
    const float*     W2l  = (const float*)d_in[5];
    const float*     b2   = (const float*)d_in[6];
    const float*     W2r  = (const float*)d_in[7];
    float*           out  = (float*)d_out;

    const long long* src = ei;            // edge_index[0]
    const long long* dst = ei + N_EDGES;  // edge_index[1]

    // Workspace layout (~103 MB): rdeg | agg | h
    char* ws = (char*)d_ws;
    float* rdeg = (float*)(ws);                                   // 100000 f
    float* agg  = (float*)(ws + (1u << 20));                      // N*128 f
    float* h    = (float*)(ws + (1u << 20) + (size_t)N_NODES * D * 4 + 1024);
    (void)ws_size; (void)in_sizes; (void)n_in; (void)out_size;

    const int nfeat  = N_NODES * D;                 // 12.8M
    const int nf4    = nfeat / 4;                   // 3.2M
    const int zb4    = (nf4 + 255) / 256;
    const int degB   = (N_EDGES + 255) / 256;
    const int nodB   = (N_NODES + 255) / 256;
    const int scatB  = (int)(((long long)N_EDGES * 32 + 255) / 256);  // 200000
    const int gemmB  = N_NODES / 16;                                  // 6250

    // degree -> reciprocal (shared by both layers)
    zero_f<<<nodB, 256, 0, stream>>>(rdeg, N_NODES);
    degree_kernel<<<degB, 256, 0, stream>>>(dst, rdeg, N_EDGES);
    recip_kernel<<<nodB, 256, 0, stream>>>(rdeg, N_NODES);

    // ---- layer 1: h = relu(mean_agg(x) @ W1l + b1 + x @ W1r) ----
    zero_f4<<<zb4, 256, 0, stream>>>((float4*)agg, nf4);
    scatter_add_kernel<<<scatB, 256, 0, stream>>>(x, src, dst, agg, N_EDGES);
    sage_gemm_kernel<true><<<gemmB, 256, 0, stream>>>(agg, x, rdeg, W1l, b1, W1r, h);

    // ---- layer 2: out = mean_agg(h) @ W2l + b2 + h @ W2r ----
    zero_f4<<<zb4, 256, 0, stream>>>((float4*)agg, nf4);
    scatter_add_kernel<<<scatB, 256, 0, stream>>>(h, src, dst, agg, N_EDGES);
    sage_gemm_kernel<false><<<gemmB, 256, 0, stream>>>(agg, h, rdeg, W2l, b2, W2r, out);
}